// MultiHeadAttention2d_60241211293768
// MI455X (gfx1250) — compile-verified
//
#include <hip/hip_runtime.h>

#define C_  64
#define HW_ 1296   // 36*36, = 81 * 16

typedef __attribute__((ext_vector_type(2))) float v2f;
typedef __attribute__((ext_vector_type(8))) float v8f;

// ---------------------------------------------------------------------------
// Kernel 1: all four 1x1-conv projections as (64x64)x(64x1296) fp32 GEMMs
// using V_WMMA_F32_16X16X4_F32. One wave32 per 16x16 output tile, 16 chained
// K=4 WMMAs. grid = (81 n-tiles, 4 m-tiles, 4 projections), block = 32.
// ---------------------------------------------------------------------------
__global__ __launch_bounds__(32) void proj_wmma_kernel(
    const float* __restrict__ x,  const float* __restrict__ y,
    const float* __restrict__ Wq, const float* __restrict__ bq,
    const float* __restrict__ Wk, const float* __restrict__ bk,
    const float* __restrict__ Wv, const float* __restrict__ bv,
    const float* __restrict__ Wp,
    float* __restrict__ q, float* __restrict__ k,
    float* __restrict__ v, float* __restrict__ yp)
{
    const int lane = threadIdx.x;          // 0..31, wave32
    const int n0   = blockIdx.x * 16;      // output column tile
    const int m0   = blockIdx.y * 16;      // output row tile

    const float* W; const float* B; const float* S; float* D;
    switch (blockIdx.z) {                  // uniform per block -> EXEC stays all-1s
      case 0:  W = Wq; B = bq;      S = x; D = q;  break;
      case 1:  W = Wk; B = bk;      S = x; D = k;  break;
      case 2:  W = Wv; B = bv;      S = y; D = v;  break;
      default: W = Wp; B = nullptr; S = y; D = yp; break;
    }

    // A-matrix 16x4 f32 layout (ISA 7.12.2): lanes 0-15 hold K=0,1; lanes
    // 16-31 hold K=2,3; M = lane & 15 across both halves.
    const int mrow  = m0 + (lane & 15);
    const int khalf = (lane >> 4) << 1;    // 0 or 2
    // B-matrix 4x16 f32: rows striped across lanes within a VGPR:
    // VGPR0 -> K = khalf, VGPR1 -> K = khalf+1, N = lane & 15.
    const int ncol  = n0 + (lane & 15);

    v8f acc = {};
    #pragma unroll
    for (int k0 = 0; k0 < C_; k0 += 4) {
        v2f a, b;
        a.x = W[mrow * C_ + k0 + khalf];
        a.y = W[mrow * C_ + k0 + khalf + 1];
        b.x = S[(k0 + khalf)     * HW_ + ncol];
        b.y = S[(k0 + khalf + 1) * HW_ + ncol];
        // 8 args: (neg_a, A, neg_b, B, c_mod, C, reuse_a, reuse_b)
        acc = __builtin_amdgcn_wmma_f32_16x16x4_f32(
            false, a, false, b, (short)0, acc, false, false);
    }

    // C/D 16x16 f32 layout: lane<16 -> M = r, lane>=16 -> M = r+8 (r = VGPR).
    const int rbase = m0 + ((lane >> 4) << 3);
    #pragma unroll
    for (int r = 0; r < 8; ++r) {
        float o = acc[r];
        if (B) o += B[rbase + r];
        D[(rbase + r) * HW_ + ncol] = o;
    }
}

// ---------------------------------------------------------------------------
// Kernel 2: rank-1 softmax attention + gamma-gated residual blend.
// energy is rank-1 per channel, so out[c,i] = (sum_j e^{s k_j} v_j)/(sum_j
// e^{s k_j}), s = q[c,i]. k is pre-scaled by log2(e) so the inner loop is
// fmaf + v_exp_f32 + 2 accumulating FMAs. Stable via m = s*kmax or s*kmin.
// grid = (4 query blocks of 324, 64 channels), block = 256 (8 wave32s).
// ---------------------------------------------------------------------------
__global__ __launch_bounds__(256) void attn_kernel(
    const float* __restrict__ q, const float* __restrict__ k,
    const float* __restrict__ v, const float* __restrict__ yp,
    const float* __restrict__ gamma, float* __restrict__ out)
{
    __shared__ float sk[HW_];
    __shared__ float sv[HW_];
    __shared__ float smax[256];
    __shared__ float smin[256];

    const int c   = blockIdx.y;
    const int ib  = blockIdx.x;          // 0..3, 324 queries each
    const int tid = threadIdx.x;
    const float LOG2E = 1.4426950408889634f;

    __builtin_prefetch(&q[c * HW_ + ib * 324], 0, 1);   // global_prefetch_b8

    float lmax = -3.402823466e38f, lmin = 3.402823466e38f;
    for (int j = tid; j < HW_; j += 256) {
        float kj = k[c * HW_ + j] * LOG2E;
        sk[j] = kj;
        sv[j] = v[c * HW_ + j];
        lmax = fmaxf(lmax, kj);
        lmin = fminf(lmin, kj);
    }
    smax[tid] = lmax; smin[tid] = lmin;
    __syncthreads();
    #pragma unroll
    for (int sred = 128; sred > 0; sred >>= 1) {
        if (tid < sred) {
            smax[tid] = fmaxf(smax[tid], smax[tid + sred]);
            smin[tid] = fminf(smin[tid], smin[tid + sred]);
        }
        __syncthreads();
    }
    const float kmax = smax[0];
    const float kmin = smin[0];

    const float g = gamma[0];
    const float inv1pg = 1.0f / (1.0f + g);

    for (int il = tid; il < 324; il += 256) {
        const int i = ib * 324 + il;
        const float s = q[c * HW_ + i];
        const float m = (s >= 0.0f) ? s * kmax : s * kmin;  // max_j s*sk[j]
        float num = 0.0f, den = 0.0f;
        #pragma unroll 4
        for (int j = 0; j < HW_; ++j) {
            // all lanes read the same sk[j]/sv[j] -> LDS broadcast
            float e = exp2f(fmaf(s, sk[j], -m));
            den += e;
            num = fmaf(e, sv[j], num);
        }
        out[c * HW_ + i] = (g * (num / den) + yp[c * HW_ + i]) * inv1pg;
    }
}

// ---------------------------------------------------------------------------
extern "C" void kernel_launch(void* const* d_in, const int* in_sizes, int n_in,
                              void* d_out, int out_size, void* d_ws, size_t ws_size,
                              hipStream_t stream) {
    const float* x     = (const float*)d_in[0];
    const float* y     = (const float*)d_in[1];
    const float* Wq    = (const float*)d_in[2];
    const float* bq    = (const float*)d_in[3];
    const float* Wk    = (const float*)d_in[4];
    const float* bk    = (const float*)d_in[5];
    const float* Wv    = (const float*)d_in[6];
    const float* bv    = (const float*)d_in[7];
    const float* Wp    = (const float*)d_in[8];
    const float* gamma = (const float*)d_in[9];

    // workspace: q, k, v, yproj -- 4 * 64 * 1296 floats = ~1.33 MB
    float* ws = (float*)d_ws;
    float* q  = ws;
    float* k  = ws + 1 * C_ * HW_;
    float* v  = ws + 2 * C_ * HW_;
    float* yp = ws + 3 * C_ * HW_;

    dim3 g1(HW_ / 16, C_ / 16, 4);   // 81 x 4 x 4 tiles, 1 wave each
    proj_wmma_kernel<<<g1, 32, 0, stream>>>(x, y, Wq, bq, Wk, bk, Wv, bv, Wp,
                                            q, k, v, yp);

    dim3 g2(4, C_);                  // 4 query blocks x 64 channels
    attn_kernel<<<g2, 256, 0, stream>>>(q, k, v, yp, gamma, (float*)d_out);
}